// CrossHeadOnlineHadamardHook_52510270161579
// MI455X (gfx1250) — compile-verified
//
#include <hip/hip_runtime.h>

// Hadamard mix across 32 heads:  y[t,h,d] = (1/sqrt(32)) * sum_k had_K[h][k] * x[t,k,d]
// had_K is symmetric (Sylvester), so per token:
//   Y^T(128x32) = X^T(128x32) * had_K(32x32) * (1/sqrt(32))
// computed on the CDNA5 matrix pipe via V_WMMA_F32_16X16X4_F32 (K=32 in 8 chunks of 4).
// B operand is loaded straight from had_K (4 KB, WGP$-resident, immediate-offset
// loads -> no address VALU); the 1/sqrt(32) scale is applied to the accumulators.

typedef __attribute__((ext_vector_type(2))) float v2f;
typedef __attribute__((ext_vector_type(4))) float v4f;
typedef __attribute__((ext_vector_type(8))) float v8f;

#define HEAD_DIM 128
#define NUM_HEADS 32
#define HIDDEN 4096  // NUM_HEADS * HEAD_DIM

__global__ __launch_bounds__(256) void cross_head_hadamard_wmma(
    const float* __restrict__ x, const float* __restrict__ hadK,
    float* __restrict__ y, int ntok) {
  const int tid  = blockIdx.x * blockDim.x + threadIdx.x;
  const int wave = tid >> 5;     // one wave = one (token, 16-row d-tile)
  const int lane = tid & 31;

  const int token = wave >> 3;   // 8 d-tiles of 16 per token (HEAD_DIM=128)
  const int d0    = (wave & 7) << 4;
  if (token >= ntok) return;     // uniform per wave; grid sized so never taken

  const int n     = lane & 15;          // N column within a 16-wide tile
  const int khalf = (lane >> 4) << 1;   // A/B K-sublayout: lanes 0-15 -> K0/K1, 16-31 -> K2/K3

  // ---- B operand: had_K rows k = 4kc+khalf+j, cols h = n (tile 0) / 16+n (tile 1).
  // Base pointer carries all lane-dependence; per-element offsets are immediates.
  const float* bp = hadK + (size_t)(khalf * NUM_HEADS) + n;
  v2f b0[8], b1[8];
#pragma unroll
  for (int kc = 0; kc < 8; ++kc) {
    b0[kc][0] = bp[(4 * kc + 0) * NUM_HEADS];
    b0[kc][1] = bp[(4 * kc + 1) * NUM_HEADS];
    b1[kc][0] = bp[(4 * kc + 0) * NUM_HEADS + 16];
    b1[kc][1] = bp[(4 * kc + 1) * NUM_HEADS + 16];
  }

  // ---- A operand: X^T rows d0..d0+15, all 32 k (streamed, read exactly once).
  // A 16x4 f32 layout: lanes 0-15 M=lane (K0 in v0, K1 in v1); lanes 16-31 K2/K3.
  const float* xb = x + (size_t)token * HIDDEN + (size_t)d0 + n +
                    (size_t)khalf * HEAD_DIM;
  v2f a[8];
#pragma unroll
  for (int kc = 0; kc < 8; ++kc) {
    a[kc][0] = xb[(4 * kc + 0) * HEAD_DIM];
    a[kc][1] = xb[(4 * kc + 1) * HEAD_DIM];
  }

  // ---- 16 accumulating WMMAs: K=32 in chunks of 4, two N-tiles (h<16, h>=16) ----
  v8f c0 = {};
  v8f c1 = {};
#pragma unroll
  for (int kc = 0; kc < 8; ++kc) {
    c0 = __builtin_amdgcn_wmma_f32_16x16x4_f32(
        /*neg_a=*/false, a[kc], /*neg_b=*/false, b0[kc],
        /*c_mod=*/(short)0, c0, /*reuse_a=*/false, /*reuse_b=*/false);
    c1 = __builtin_amdgcn_wmma_f32_16x16x4_f32(
        false, a[kc], false, b1[kc], (short)0, c1, false, false);
  }

  // ---- Apply 1/sqrt(32) scale on accumulators (packed/dual-issued muls) ----
  const float s = 0.17677669529663687f;
#pragma unroll
  for (int i = 0; i < 8; ++i) {
    c0[i] *= s;
    c1[i] *= s;
  }

  // ---- Store: D vgpr v holds d = d0+v (lanes 0-15) / d0+8+v (lanes 16-31),
  //      column h = h0 + n  ->  each lane owns 8 consecutive d's: two b128 stores
  //      per N-tile. Output is write-once: non-temporal.
  const int dlo = d0 + ((lane >> 4) << 3);
  float* y0 = y + (size_t)token * HIDDEN + (size_t)n * HEAD_DIM + dlo;  // h-tile 0
  float* y1 = y0 + (size_t)16 * HEAD_DIM;                               // h-tile 1

  v4f t;
  t[0] = c0[0]; t[1] = c0[1]; t[2] = c0[2]; t[3] = c0[3];
  __builtin_nontemporal_store(t, (v4f*)(y0 + 0));
  t[0] = c0[4]; t[1] = c0[5]; t[2] = c0[6]; t[3] = c0[7];
  __builtin_nontemporal_store(t, (v4f*)(y0 + 4));
  t[0] = c1[0]; t[1] = c1[1]; t[2] = c1[2]; t[3] = c1[3];
  __builtin_nontemporal_store(t, (v4f*)(y1 + 0));
  t[0] = c1[4]; t[1] = c1[5]; t[2] = c1[6]; t[3] = c1[7];
  __builtin_nontemporal_store(t, (v4f*)(y1 + 4));
}

extern "C" void kernel_launch(void* const* d_in, const int* in_sizes, int n_in,
                              void* d_out, int out_size, void* d_ws, size_t ws_size,
                              hipStream_t stream) {
  const float* x    = (const float*)d_in[0];
  const float* hadK = (const float*)d_in[1];  // 32x32 row-major, WGP$-resident
  float* out = (float*)d_out;

  const int ntok = in_sizes[0] / HIDDEN;  // B*S = 16384
  // 8 waves per token (one per 16-row d-tile), 8 waves per 256-thread block
  // -> one block per token.
  cross_head_hadamard_wmma<<<ntok, 256, 0, stream>>>(x, hadK, out, ntok);
}